// PDFormer_82703890252057
// MI455X (gfx1250) — compile-verified
//
#include <hip/hip_runtime.h>
#include <math.h>
#include <stdint.h>

// ---------------------------------------------------------------------------
// Problem constants (match reference)
// ---------------------------------------------------------------------------
constexpr int Bc = 4, Tc = 12, Nc = 512, Dc = 64;
constexpr int GHc = 4, SHc = 2, THc = 2, HDc = 8;
constexpr int GDc = 32, SDc = 16, TDc = 16;
constexpr int Kpat = 16;
constexpr int BT   = Bc * Tc;           // 48
constexpr int MTOT = BT * Nc;           // 24576 tokens
constexpr float SCALE_F = 0.35355339059327373f;  // 8^-0.5

typedef __attribute__((ext_vector_type(2))) float v2f;
typedef __attribute__((ext_vector_type(8))) float v8f;

// f32 WMMA: D(16x16,f32) = A(16x4,f32) x B(4x16,f32) + C
// (confirmed lowering: v_wmma_f32_16x16x4_f32)
__device__ __forceinline__ v8f wmma_f32x4(v2f a, v2f b, v8f c) {
  return __builtin_amdgcn_wmma_f32_16x16x4_f32(
      false, a, false, b, (short)0, c, false, false);
}

// ---------------------------------------------------------------------------
// Generic 1-wave WMMA GEMM: C[M,Nout] = A[M,K] @ W[K,Nout] + bias
// grid = (M/16, Nout/16), block = 32 (one wave, EXEC all ones).
// KDIM is a compile-time constant -> K-loop fully unrolls into KDIM/4
// back-to-back v_wmma_f32_16x16x4_f32 with prefetched operand loads.
//
// Fragment layouts (CDNA5 ISA 7.12.2):
//  A 16x4 : lane L -> row M=L%16 ; VGPR0/1 hold K = {0,1} (L<16) or {2,3}
//  B 4x16 : lane L -> col N=L%16 ; VGPR0/1 hold K = {0,1} (L<16) or {2,3}
//  C 16x16: lane L, VGPR v -> row v + (L<16?0:8), col L%16
// ---------------------------------------------------------------------------
template <int KDIM>
__global__ __launch_bounds__(32)
void gemm_wmma_f32(const float* __restrict__ A, int lda,
                   const float* __restrict__ W, int ldw,
                   const float* __restrict__ bias,
                   float* __restrict__ C, int ldc) {
  const int lane = threadIdx.x;
  const int lm   = lane & 15;
  const int hi   = lane >> 4;       // 0 or 1
  const int kofs = hi * 2;
  const int col  = blockIdx.y * 16 + lm;

  const float* Arow = A + (size_t)(blockIdx.x * 16 + lm) * lda;
  __builtin_prefetch(W + (size_t)kofs * ldw + col, 0, 0);  // global_prefetch_b8

  v8f acc = {};
#pragma unroll
  for (int k0 = 0; k0 < KDIM; k0 += 4) {
    v2f a, b;
    a.x = Arow[k0 + kofs];
    a.y = Arow[k0 + kofs + 1];
    b.x = W[(size_t)(k0 + kofs    ) * ldw + col];
    b.y = W[(size_t)(k0 + kofs + 1) * ldw + col];
    acc = wmma_f32x4(a, b, acc);
  }
  const float bv = bias ? bias[col] : 0.0f;
#pragma unroll
  for (int v = 0; v < 8; ++v) {
    const int row = blockIdx.x * 16 + v + hi * 8;
    C[(size_t)row * ldc + col] = acc[v] + bv;
  }
}

// ---------------------------------------------------------------------------
// Pattern-key attention (geo-key enrichment), O=1:
//   pk = pattern_keys @ pat_kw + pat_kb          [16,32]
//   pv = pattern_keys @ pat_vw + pat_vb          [16,32]
//   pa = softmax(pq @ pk^T * scale)              per token over 16 keys
//   gk[token] += pa @ pv
// K=16 is far below a 16x16 tile's efficiency point -> VALU.
// ---------------------------------------------------------------------------
__global__ __launch_bounds__(256)
void pattern_attn(const float* __restrict__ pq,             // [MTOT,32]
                  const float* __restrict__ pattern_keys,   // [16,64]
                  const float* __restrict__ pat_kw,         // [64,32]
                  const float* __restrict__ pat_kb,         // [32]
                  const float* __restrict__ pat_vw,         // [64,32]
                  const float* __restrict__ pat_vb,         // [32]
                  float* __restrict__ gk) {                 // [MTOT,32] in-place
  __shared__ float sPK[Kpat][GDc];
  __shared__ float sPV[Kpat][GDc];
  const int tid = threadIdx.x;
  for (int i = tid; i < Kpat * GDc; i += 256) {
    const int kk = i >> 5, c = i & 31;
    float aK = pat_kb[c], aV = pat_vb[c];
    for (int d = 0; d < Dc; ++d) {
      const float pkv = pattern_keys[kk * Dc + d];
      aK += pkv * pat_kw[d * GDc + c];
      aV += pkv * pat_vw[d * GDc + c];
    }
    sPK[kk][c] = aK;
    sPV[kk][c] = aV;
  }
  __syncthreads();

  const int tok = blockIdx.x * 256 + tid;
  if (tok >= MTOT) return;

  float q[GDc];
#pragma unroll
  for (int c = 0; c < GDc; ++c) q[c] = pq[(size_t)tok * GDc + c];

  float sc[Kpat];
  float mx = -INFINITY;
#pragma unroll
  for (int kk = 0; kk < Kpat; ++kk) {
    float d = 0.0f;
#pragma unroll
    for (int c = 0; c < GDc; ++c) d += q[c] * sPK[kk][c];
    sc[kk] = d * SCALE_F;
    mx = fmaxf(mx, sc[kk]);
  }
  float sum = 0.0f;
#pragma unroll
  for (int kk = 0; kk < Kpat; ++kk) { sc[kk] = __expf(sc[kk] - mx); sum += sc[kk]; }
  const float inv = 1.0f / sum;
#pragma unroll
  for (int c = 0; c < GDc; ++c) {
    float o = 0.0f;
#pragma unroll
    for (int kk = 0; kk < Kpat; ++kk) o += sc[kk] * sPV[kk][c];
    gk[(size_t)tok * GDc + c] += o * inv;
  }
}

// ---------------------------------------------------------------------------
// Temporal attention: per (b, n, h), softmax(Q K^T / sqrt(hd)) V over T=12.
// T=12 is too small for 16x16 WMMA tiles (44% pad waste); 49152 independent
// threads keep the VALU busy instead. Writes cat[:, 0:16].
// ---------------------------------------------------------------------------
__global__ __launch_bounds__(256)
void temporal_attn(const float* __restrict__ tq,   // [MTOT,16]
                   const float* __restrict__ tk,
                   const float* __restrict__ tv,
                   float* __restrict__ cat) {      // [MTOT,64]
  const int idx = blockIdx.x * 256 + threadIdx.x;  // (((b*N+n)*TH+h)*T+t)
  const int t = idx % Tc;
  int r = idx / Tc;
  const int h = r % THc;  r /= THc;
  const int n = r % Nc;
  const int b = r / Nc;
  if (b >= Bc) return;

  const int ch = h * HDc;
  const size_t tokQ = ((size_t)(b * Tc + t) * Nc + n);

  float q[HDc];
#pragma unroll
  for (int d = 0; d < HDc; ++d) q[d] = tq[tokQ * TDc + ch + d];

  float sc[Tc];
  float mx = -INFINITY;
  for (int s = 0; s < Tc; ++s) {
    const size_t tokS = ((size_t)(b * Tc + s) * Nc + n);
    float dot = 0.0f;
#pragma unroll
    for (int d = 0; d < HDc; ++d) dot += q[d] * tk[tokS * TDc + ch + d];
    sc[s] = dot * SCALE_F;
    mx = fmaxf(mx, sc[s]);
  }
  float sum = 0.0f;
  for (int s = 0; s < Tc; ++s) { sc[s] = __expf(sc[s] - mx); sum += sc[s]; }
  const float inv = 1.0f / sum;

  float o[HDc] = {};
  for (int s = 0; s < Tc; ++s) {
    const size_t tokS = ((size_t)(b * Tc + s) * Nc + n);
    const float p = sc[s] * inv;
#pragma unroll
    for (int d = 0; d < HDc; ++d) o[d] += p * tv[tokS * TDc + ch + d];
  }
#pragma unroll
  for (int d = 0; d < HDc; ++d) cat[tokQ * Dc + ch + d] = o[d];  // t_x slot [0,16)
}

// ---------------------------------------------------------------------------
// Spatial attention (geo and sem branches): per (b,t,h), N=512, hd=8, with
// boolean mask -> -inf before softmax. One wave per 16-row tile:
//   V staged memory->LDS with GLOBAL_LOAD_ASYNC_TO_LDS_B128 (ASYNCcnt),
//     latency hidden behind the score + softmax phases
//   scores tile (16x16) = 2 x WMMA_F32_16X16X4 (K=hd=8)
//   full-row softmax through LDS (row pitch 516 -> conflict-free col access)
//   out = P @ V : 128 x WMMA_F32_16X16X4 (K=512), V zero-padded to 16 cols
// grid = (N/16, B*T, H), block = 32.
// ---------------------------------------------------------------------------
__global__ __launch_bounds__(32)
void spatial_attn(const float* __restrict__ qbuf,   // [MTOT,C]
                  const float* __restrict__ kbuf,
                  const float* __restrict__ vbuf,
                  int C,                            // H*8
                  const int* __restrict__ mask,     // [N,N], nonzero = masked
                  float* __restrict__ cat,          // [MTOT,64]
                  int catOfs) {
  constexpr int PITCH = Nc + 4;                     // 516: bank-conflict-free cols
  __shared__ float sP[16][PITCH];
  __shared__ __align__(16) float sV[Nc][HDc];

  const int rtile = blockIdx.x;
  const int bt    = blockIdx.y;
  const int h     = blockIdx.z;
  const int lane  = threadIdx.x;
  const int lm    = lane & 15;
  const int hi    = lane >> 4;
  const int kofs  = hi * 2;
  const int ch    = h * HDc;
  const size_t base = (size_t)bt * Nc;

  // ---- async-stage V[512,8] into LDS: 1024 x 16B transfers, 32 per lane.
  // global_load_async_to_lds_b128: VDST = per-lane LDS byte address,
  // VADDR = 64-bit global address (GV mode). Tracked by ASYNCcnt; the wait
  // is deferred until just before the P@V phase so the transfers overlap
  // the QK^T WMMAs and the softmax.
  for (int j = lane; j < Nc * 2; j += 32) {
    const int row  = j >> 1;
    const int half = j & 1;                               // 4-float half-row
    const float*   gptr   = vbuf + (base + row) * C + ch + half * 4;
    const uint32_t ldsOff = (uint32_t)(uintptr_t)&sV[row][half * 4];
    asm volatile("global_load_async_to_lds_b128 %0, %1, off"
                 :: "v"(ldsOff), "v"(gptr)
                 : "memory");
  }

  // Q A-fragments (row = rtile*16 + lm), K split as k0=0 and k0=4.
  const float* qrow = qbuf + (base + rtile * 16 + lm) * C + ch;
  v2f qa0, qa1;
  qa0.x = qrow[kofs];     qa0.y = qrow[kofs + 1];
  qa1.x = qrow[4 + kofs]; qa1.y = qrow[4 + kofs + 1];

  // ---- scores: loop over 32 column tiles ----
  for (int j = 0; j < Nc / 16; ++j) {
    const float* krow = kbuf + (base + j * 16 + lm) * C + ch;  // col = j*16+lm
    v2f kb0, kb1;                                              // B[d][col] = K[col][d]
    kb0.x = krow[kofs];     kb0.y = krow[kofs + 1];
    kb1.x = krow[4 + kofs]; kb1.y = krow[4 + kofs + 1];

    v8f c = {};
    c = wmma_f32x4(qa0, kb0, c);
    c = wmma_f32x4(qa1, kb1, c);

#pragma unroll
    for (int v = 0; v < 8; ++v) {
      const int rloc = v + hi * 8;
      const int rowg = rtile * 16 + rloc;
      const int colg = j * 16 + lm;
      float s = c[v] * SCALE_F;
      if (mask[(size_t)rowg * Nc + colg]) s = -INFINITY;
      sP[rloc][colg] = s;
    }
  }
  __syncthreads();

  // ---- softmax over 512 columns, all 32 lanes cooperate per row ----
  for (int r = 0; r < 16; ++r) {
    float m = -INFINITY;
    for (int c = lane; c < Nc; c += 32) m = fmaxf(m, sP[r][c]);
#pragma unroll
    for (int off = 16; off > 0; off >>= 1) m = fmaxf(m, __shfl_xor(m, off, 32));
    float sum = 0.0f;
    for (int c = lane; c < Nc; c += 32) {
      const float e = __expf(sP[r][c] - m);
      sP[r][c] = e;
      sum += e;
    }
#pragma unroll
    for (int off = 16; off > 0; off >>= 1) sum += __shfl_xor(sum, off, 32);
    const float inv = 1.0f / sum;
    for (int c = lane; c < Nc; c += 32) sP[r][c] *= inv;
  }
  __syncthreads();

  // All async V transfers must have landed in LDS before the P@V reads.
  asm volatile("s_wait_asynccnt 0x0" ::: "memory");

  // ---- out = P[16,512] @ V[512,8 (padded to 16)] ----
  v8f acc = {};
#pragma unroll 4
  for (int k0 = 0; k0 < Nc; k0 += 4) {
    v2f a, b;
    a.x = sP[lm][k0 + kofs];
    a.y = sP[lm][k0 + kofs + 1];
    b.x = (lm < HDc) ? sV[k0 + kofs][lm]     : 0.0f;
    b.y = (lm < HDc) ? sV[k0 + kofs + 1][lm] : 0.0f;
    acc = wmma_f32x4(a, b, acc);
  }
  if (lm < HDc) {
#pragma unroll
    for (int v = 0; v < 8; ++v) {
      const int rowg = rtile * 16 + v + hi * 8;
      cat[(base + rowg) * Dc + catOfs + ch + lm] = acc[v];
    }
  }
}

// ---------------------------------------------------------------------------
// Host-side orchestration
// ---------------------------------------------------------------------------
extern "C" void kernel_launch(void* const* d_in, const int* in_sizes, int n_in,
                              void* d_out, int out_size, void* d_ws, size_t ws_size,
                              hipStream_t stream) {
  (void)in_sizes; (void)n_in; (void)out_size; (void)ws_size;

  const float* x        = (const float*)d_in[0];   // [B,T,N,D]
  const float* x_pat    = (const float*)d_in[1];   // [B,T,N,D,1] == [MTOT,64]
  const float* pkeys    = (const float*)d_in[2];   // [16,64,1]  == [16,64]
  const float* t_qw  = (const float*)d_in[3];  const float* t_qb  = (const float*)d_in[4];
  const float* t_kw  = (const float*)d_in[5];  const float* t_kb  = (const float*)d_in[6];
  const float* t_vw  = (const float*)d_in[7];  const float* t_vb  = (const float*)d_in[8];
  const float* g_qw  = (const float*)d_in[9];  const float* g_qb  = (const float*)d_in[10];
  const float* g_kw  = (const float*)d_in[11]; const float* g_kb  = (const float*)d_in[12];
  const float* g_vw  = (const float*)d_in[13]; const float* g_vb  = (const float*)d_in[14];
  const float* s_qw  = (const float*)d_in[15]; const float* s_qb  = (const float*)d_in[16];
  const float* s_kw  = (const float*)d_in[17]; const float* s_kb  = (const float*)d_in[18];
  const float* s_vw  = (const float*)d_in[19]; const float* s_vb  = (const float*)d_in[20];
  const float* p_qw  = (const float*)d_in[21]; const float* p_qb  = (const float*)d_in[22];
  const float* p_kw  = (const float*)d_in[23]; const float* p_kb  = (const float*)d_in[24];
  const float* p_vw  = (const float*)d_in[25]; const float* p_vb  = (const float*)d_in[26];
  const float* proj_w = (const float*)d_in[27]; const float* proj_b = (const float*)d_in[28];
  const int*   geo_mask = (const int*)d_in[29];   // bool -> integer contract
  const int*   sem_mask = (const int*)d_in[30];

  // Workspace layout (floats), ~28.3 MB total
  float* ws = (float*)d_ws;
  const size_t M = MTOT;
  float* tq  = ws;
  float* tk  = tq + M * TDc;
  float* tv  = tk + M * TDc;
  float* gq  = tv + M * TDc;
  float* gk  = gq + M * GDc;
  float* gv  = gk + M * GDc;
  float* sq  = gv + M * GDc;
  float* sk  = sq + M * SDc;
  float* sv  = sk + M * SDc;
  float* pq  = sv + M * SDc;
  float* cat = pq + M * GDc;

  auto gemm = [&](const float* A, const float* W, const float* bias,
                  float* C, int Nout) {
    dim3 g(MTOT / 16, Nout / 16);
    gemm_wmma_f32<Dc><<<g, 32, 0, stream>>>(A, Dc, W, Nout, bias, C, Nout);
  };

  // 1) All input projections (WMMA GEMMs, K=64, fully unrolled)
  gemm(x,     t_qw, t_qb, tq, TDc);
  gemm(x,     t_kw, t_kb, tk, TDc);
  gemm(x,     t_vw, t_vb, tv, TDc);
  gemm(x,     g_qw, g_qb, gq, GDc);
  gemm(x,     g_kw, g_kb, gk, GDc);
  gemm(x,     g_vw, g_vb, gv, GDc);
  gemm(x,     s_qw, s_qb, sq, SDc);
  gemm(x,     s_kw, s_kb, sk, SDc);
  gemm(x,     s_vw, s_vb, sv, SDc);
  gemm(x_pat, p_qw, p_qb, pq, GDc);   // pattern queries (O=1)

  // 2) Pattern attention: gk += softmax(pq pk^T) pv
  pattern_attn<<<MTOT / 256, 256, 0, stream>>>(pq, pkeys, p_kw, p_kb, p_vw, p_vb, gk);

  // 3) Temporal branch -> cat[:, 0:16]
  temporal_attn<<<(Bc * Nc * THc * Tc) / 256, 256, 0, stream>>>(tq, tk, tv, cat);

  // 4) Geo spatial attention -> cat[:, 16:48]
  spatial_attn<<<dim3(Nc / 16, BT, GHc), 32, 0, stream>>>(gq, gk, gv, GDc,
                                                          geo_mask, cat, TDc);
  // 5) Sem spatial attention -> cat[:, 48:64]
  spatial_attn<<<dim3(Nc / 16, BT, SHc), 32, 0, stream>>>(sq, sk, sv, SDc,
                                                          sem_mask, cat, TDc + GDc);

  // 6) Final projection: out = cat @ proj_w + proj_b
  gemm(cat, proj_w, proj_b, (float*)d_out, Dc);
}